// DepthTerm_20126216749838
// MI455X (gfx1250) — compile-verified
//
#include <hip/hip_runtime.h>
#include <hip/hip_bf16.h>

// ICP depth-term on gfx1250:
//  - v_wmma_f32_16x16x4_f32 computes 16x16 tiles of point dots (K=3 padded to 4)
//    and normal dots; d2 = q2 + r2 - 2*dot via fma.
//  - pass 1: per-query overall-min d2 and valid-min d2 (valid = d2<0.0025 && cos>cos15deg)
//  - pass 2: rank-count of valid-min over all refs; chosen = rank<32 ? valid-min : overall-min
//  - deterministic tree reduction of sqrt(chosen)/n into the scalar output.

typedef __attribute__((ext_vector_type(2))) float v2f;
typedef __attribute__((ext_vector_type(8))) float v8f;

#define KNN_K     32
#define DIST_TH2  0.0025f                   // 0.05^2
#define COS_TH    0.9659258262890683f       // cos(pi/12)
#define BIGF      1e30f

// A/B fragment for V_WMMA_F32_16X16X4_F32 (32-bit, 16x4 / 4x16):
// lanes 0-15 hold (x,y) of point i, lanes 16-31 hold (z,0) of point i.
__device__ __forceinline__ v2f load_frag3(const float* __restrict__ p, int i, int half) {
  const float* q = p + 3 * i;
  v2f f;
  f.x = half ? q[2] : q[0];
  f.y = half ? 0.0f : q[1];
  return f;
}

__global__ void sqnorm_kernel(const float* __restrict__ pts, float* __restrict__ out, int n) {
  int i = blockIdx.x * blockDim.x + threadIdx.x;
  if (i < n) {
    float x = pts[3 * i], y = pts[3 * i + 1], z = pts[3 * i + 2];
    out[i] = x * x + y * y + z * z;
  }
}

// Pass 1: per 16-query tile (one wave), stream all ref tiles, track
// overall-min d2 and valid-min d2 per query row.
__global__ __launch_bounds__(256) void icp_min_kernel(
    const float* __restrict__ q_pts, const float* __restrict__ q_nrm,
    const float* __restrict__ q_sq, int nq,
    const float* __restrict__ r_pts, const float* __restrict__ r_nrm,
    const float* __restrict__ r_sq, int nr,
    float* __restrict__ min_all, float* __restrict__ min_valid) {
  const int lane = threadIdx.x & 31;
  const int wave = threadIdx.x >> 5;
  const int tile = blockIdx.x * (blockDim.x >> 5) + wave;
  const int ntiles = (nq + 15) >> 4;
  if (tile >= ntiles) return;           // wave-uniform: EXEC full below (WMMA-safe)
  const int qbase = tile << 4;
  const int half = lane >> 4;
  const int l = lane & 15;

  int qi = qbase + l; if (qi >= nq) qi = nq - 1;
  const v2f a  = load_frag3(q_pts, qi, half);
  const v2f an = load_frag3(q_nrm, qi, half);

  float q2[8], mall[8], mval[8];
#pragma unroll
  for (int j = 0; j < 8; ++j) {
    int m = qbase + j + half * 8; if (m >= nq) m = nq - 1;
    q2[j] = q_sq[m];
    mall[j] = BIGF;
    mval[j] = BIGF;
  }

  const int rtiles = (nr + 15) >> 4;
  const v8f zc = {0.f, 0.f, 0.f, 0.f, 0.f, 0.f, 0.f, 0.f};
  for (int t = 0; t < rtiles; ++t) {
    const int rbase = t << 4;
    int ri = rbase + l;
    const bool pad = ri >= nr; if (pad) ri = nr - 1;
    const v2f b  = load_frag3(r_pts, ri, half);
    const v2f bn = load_frag3(r_nrm, ri, half);
    const float r2 = pad ? BIGF : r_sq[ri];

    v8f dot  = __builtin_amdgcn_wmma_f32_16x16x4_f32(false, a,  false, b,  (short)0, zc, false, false);
    v8f ndot = __builtin_amdgcn_wmma_f32_16x16x4_f32(false, an, false, bn, (short)0, zc, false, false);

#pragma unroll
    for (int j = 0; j < 8; ++j) {
      float d2 = fmaf(dot[j], -2.0f, q2[j] + r2);
      bool valid = (d2 < DIST_TH2) && (ndot[j] > COS_TH);
      mall[j] = fminf(mall[j], d2);
      mval[j] = fminf(mval[j], valid ? d2 : BIGF);
    }
  }

  // min-reduce across the 16 lanes of each half (each VGPR j == one matrix row)
#pragma unroll
  for (int j = 0; j < 8; ++j) {
#pragma unroll
    for (int m = 1; m <= 8; m <<= 1) {
      mall[j] = fminf(mall[j], __shfl_xor(mall[j], m, 32));
      mval[j] = fminf(mval[j], __shfl_xor(mval[j], m, 32));
    }
  }
  if (l == 0) {
#pragma unroll
    for (int j = 0; j < 8; ++j) {
      int m = qbase + j + half * 8;
      if (m < nq) { min_all[m] = mall[j]; min_valid[m] = mval[j]; }
    }
  }
}

// Pass 2: rank-count of valid-min d2 over all refs; emit per-tile partial sum of
// sqrt(chosen d2). Deterministic (no float atomics).
__global__ __launch_bounds__(256) void icp_count_kernel(
    const float* __restrict__ q_pts, const float* __restrict__ q_sq, int nq,
    const float* __restrict__ r_pts, const float* __restrict__ r_sq, int nr,
    const float* __restrict__ min_all, const float* __restrict__ min_valid,
    float* __restrict__ partial) {
  const int lane = threadIdx.x & 31;
  const int wave = threadIdx.x >> 5;
  const int tile = blockIdx.x * (blockDim.x >> 5) + wave;
  const int ntiles = (nq + 15) >> 4;
  if (tile >= ntiles) return;
  const int qbase = tile << 4;
  const int half = lane >> 4;
  const int l = lane & 15;

  int qi = qbase + l; if (qi >= nq) qi = nq - 1;
  const v2f a = load_frag3(q_pts, qi, half);

  float q2[8], th[8];
  int cnt[8];
#pragma unroll
  for (int j = 0; j < 8; ++j) {
    int m = qbase + j + half * 8; if (m >= nq) m = nq - 1;
    q2[j] = q_sq[m];
    th[j] = min_valid[m];
    cnt[j] = 0;
  }

  const int rtiles = (nr + 15) >> 4;
  const v8f zc = {0.f, 0.f, 0.f, 0.f, 0.f, 0.f, 0.f, 0.f};
  for (int t = 0; t < rtiles; ++t) {
    const int rbase = t << 4;
    int ri = rbase + l;
    const bool pad = ri >= nr; if (pad) ri = nr - 1;
    const v2f b = load_frag3(r_pts, ri, half);
    const float r2 = pad ? BIGF : r_sq[ri];

    v8f dot = __builtin_amdgcn_wmma_f32_16x16x4_f32(false, a, false, b, (short)0, zc, false, false);

#pragma unroll
    for (int j = 0; j < 8; ++j) {
      float d2 = fmaf(dot[j], -2.0f, q2[j] + r2);
      cnt[j] += (d2 < th[j]) ? 1 : 0;
    }
  }

  // sum-reduce rank counts across the 16 lanes of each half
#pragma unroll
  for (int j = 0; j < 8; ++j) {
#pragma unroll
    for (int m = 1; m <= 8; m <<= 1) {
      cnt[j] += __shfl_xor(cnt[j], m, 32);
    }
  }

  float local = 0.0f;
  if (l == 0) {
#pragma unroll
    for (int j = 0; j < 8; ++j) {
      int m = qbase + j + half * 8;
      if (m < nq) {
        float mv = min_valid[m];
        float ma = min_all[m];
        // candidate picked iff it exists and lies within the 32 nearest
        float chosen = (mv < 1e29f && cnt[j] < KNN_K) ? mv : ma;
        local += sqrtf(fmaxf(chosen, 0.0f));
      }
    }
  }
  local += __shfl_xor(local, 16, 32);   // fold lane16 (rows 8-15) into lane0
  if (lane == 0) partial[tile] = local;
}

// Deterministic final reduction: out = sum(p1)*inv1 + sum(p2)*inv2
__global__ __launch_bounds__(256) void finalize_kernel(
    const float* __restrict__ p1, int n1, float inv1,
    const float* __restrict__ p2, int n2, float inv2,
    float* __restrict__ out) {
  __shared__ float sm[256];
  float s = 0.0f;
  for (int i = threadIdx.x; i < n1; i += 256) s += p1[i];
  s *= inv1;
  float s2 = 0.0f;
  for (int i = threadIdx.x; i < n2; i += 256) s2 += p2[i];
  s = fmaf(s2, inv2, s);
  sm[threadIdx.x] = s;
  __syncthreads();
  for (int w = 128; w > 0; w >>= 1) {
    if (threadIdx.x < w) sm[threadIdx.x] += sm[threadIdx.x + w];
    __syncthreads();
  }
  if (threadIdx.x == 0) out[0] = sm[0];
}

extern "C" void kernel_launch(void* const* d_in, const int* in_sizes, int n_in,
                              void* d_out, int out_size, void* d_ws, size_t ws_size,
                              hipStream_t stream) {
  const float* depth_v = (const float*)d_in[0];
  const float* depth_n = (const float*)d_in[1];
  const float* verts   = (const float*)d_in[2];
  const float* vnorm   = (const float*)d_in[3];
  const int M = in_sizes[0] / 3;   // 50000 depth points
  const int N = in_sizes[2] / 3;   // 6890 SMPL vertices
  const int t_s2d = (N + 15) / 16;
  const int t_d2s = (M + 15) / 16;

  float* ws      = (float*)d_ws;
  float* sq_d    = ws;                 // [M]
  float* sq_v    = sq_d + M;           // [N]
  float* ma_s2d  = sq_v + N;           // [N]
  float* mv_s2d  = ma_s2d + N;         // [N]
  float* ma_d2s  = mv_s2d + N;         // [M]
  float* mv_d2s  = ma_d2s + M;         // [M]
  float* p_s2d   = mv_d2s + M;         // [t_s2d]
  float* p_d2s   = p_s2d + t_s2d;      // [t_d2s]

  sqnorm_kernel<<<(M + 255) / 256, 256, 0, stream>>>(depth_v, sq_d, M);
  sqnorm_kernel<<<(N + 255) / 256, 256, 0, stream>>>(verts, sq_v, N);

  const int WPB = 8;  // wave32s per block
  icp_min_kernel<<<(t_s2d + WPB - 1) / WPB, WPB * 32, 0, stream>>>(
      verts, vnorm, sq_v, N, depth_v, depth_n, sq_d, M, ma_s2d, mv_s2d);
  icp_min_kernel<<<(t_d2s + WPB - 1) / WPB, WPB * 32, 0, stream>>>(
      depth_v, depth_n, sq_d, M, verts, vnorm, sq_v, N, ma_d2s, mv_d2s);

  icp_count_kernel<<<(t_s2d + WPB - 1) / WPB, WPB * 32, 0, stream>>>(
      verts, sq_v, N, depth_v, sq_d, M, ma_s2d, mv_s2d, p_s2d);
  icp_count_kernel<<<(t_d2s + WPB - 1) / WPB, WPB * 32, 0, stream>>>(
      depth_v, sq_d, M, verts, sq_v, N, ma_d2s, mv_d2s, p_d2s);

  finalize_kernel<<<1, 256, 0, stream>>>(p_s2d, t_s2d, 1.0f / (float)N,
                                         p_d2s, t_d2s, 1.0f / (float)M,
                                         (float*)d_out);
}